// LSTMEncoder_81183471829709
// MI455X (gfx1250) — compile-verified
//
#include <hip/hip_runtime.h>
#include <hip/hip_bf16.h>
#include <math.h>

typedef float v2f __attribute__((ext_vector_type(2)));
typedef float v8f __attribute__((ext_vector_type(8)));

#define Bsz 16
#define Lsz 64
#define Hsz 400
#define G4H 1600
#define XSTRIDE 800
#define FEATPAD 112

__device__ __forceinline__ float sigmoidf_(float x) { return 1.0f / (1.0f + __expf(-x)); }

// ------------------------------------------------------------------
// Embedding gather: x[b,l,0:300]=w_emb[words], x[b,l,300:400]=t_emb[pos]
// ------------------------------------------------------------------
__global__ void embed_kernel(const int* __restrict__ words, const int* __restrict__ pos,
                             const float* __restrict__ wemb, const float* __restrict__ temb,
                             float* __restrict__ x) {
  int idx = blockIdx.x * blockDim.x + threadIdx.x;
  const int total = Bsz * Lsz * 400;
  if (idx >= total) return;
  int d  = idx % 400;
  int bl = idx / 400;
  float v;
  if (d < 300) v = wemb[(size_t)words[bl] * 300 + d];
  else         v = temb[(size_t)pos[bl]   * 100 + (d - 300)];
  x[(size_t)bl * XSTRIDE + d] = v;
}

// ------------------------------------------------------------------
// Input projection: gates[dir,b,l,:] = x[b,l,:] @ Wih[dir]^T + bih + bhh
// One wave per block; each wave computes a 16(M) x 64(N) tile strip,
// reusing the A fragment across 4 N-tiles. f32 WMMA 16x16x4.
// ------------------------------------------------------------------
__global__ void lstm_input_proj(const float* __restrict__ x,    // (B*L, XSTRIDE), D valid dims
                                const float* __restrict__ Wih,  // (2, 1600, D)
                                const float* __restrict__ bih,  // (2, 1600)
                                const float* __restrict__ bhh,  // (2, 1600)
                                float* __restrict__ gates,      // (2, B*L, 1600)
                                int D) {
  const int lane   = threadIdx.x & 31;
  const int lanelo = lane & 15;
  const int khalf  = lane >> 4;
  const int mt  = blockIdx.x;   // 0..63  row tile (16 rows of B*L)
  const int ng  = blockIdx.y;   // 0..24  group of 4 column tiles (64 cols)
  const int dir = blockIdx.z;   // 0..1
  const int row = mt * 16 + lanelo;
  const float* W    = Wih + (size_t)dir * G4H * D;
  const float* xrow = x + (size_t)row * XSTRIDE;

  v8f acc[4] = {};
  for (int k0 = 0; k0 < D; k0 += 4) {
    v2f a;
    a.x = xrow[k0 + khalf * 2 + 0];
    a.y = xrow[k0 + khalf * 2 + 1];
#pragma unroll
    for (int i = 0; i < 4; ++i) {
      int col = ng * 64 + i * 16 + lanelo;
      const float* wp = W + (size_t)col * D + k0 + khalf * 2;
      v2f bfrag;
      bfrag.x = wp[0];
      bfrag.y = wp[1];
      acc[i] = __builtin_amdgcn_wmma_f32_16x16x4_f32(false, a, false, bfrag,
                                                     (short)0, acc[i], false, false);
    }
  }
  const float* bi = bih + (size_t)dir * G4H;
  const float* bh = bhh + (size_t)dir * G4H;
  float* Gp = gates + (size_t)dir * (Bsz * Lsz * G4H);
#pragma unroll
  for (int i = 0; i < 4; ++i) {
    int col = ng * 64 + i * 16 + lanelo;
    float bias = bi[col] + bh[col];
#pragma unroll
    for (int r = 0; r < 8; ++r) {
      int m = khalf * 8 + r;
      Gp[(size_t)(mt * 16 + m) * G4H + col] = acc[i][r] + bias;
    }
  }
}

// ------------------------------------------------------------------
// Recurrence: one workgroup per direction; 25 waves, wave w owns H-slice
// j = [16w, 16w+16). Each step: 4 gate tiles via WMMA over h(LDS) x Whh^T,
// then register-local LSTM cell update (c kept in C-fragment layout).
// ------------------------------------------------------------------
__global__ __launch_bounds__(800, 1)
void lstm_recurrence(const float* __restrict__ Whh_layer, // (2, 1600, 400)
                     const float* __restrict__ gates,     // (2, B*L, 1600)  (= xs)
                     float* __restrict__ xout) {          // (B*L, XSTRIDE)
  __shared__ float hLDS[Bsz * Hsz];
  const int tid    = threadIdx.x;
  const int lane   = tid & 31;
  const int lanelo = lane & 15;
  const int khalf  = lane >> 4;
  const int wave   = tid >> 5;            // 0..24 -> H tile
  const int dir    = blockIdx.x;
  const float* Whh = Whh_layer + (size_t)dir * G4H * Hsz;
  const float* Gp  = gates + (size_t)dir * (Bsz * Lsz * G4H);

  for (int i = tid; i < Bsz * Hsz; i += blockDim.x) hLDS[i] = 0.0f;
  v8f creg = {};
  __syncthreads();

  const int j = wave * 16 + lanelo;       // output column in [0, 400)
  for (int t = 0; t < Lsz; ++t) {
    const int tt = dir ? (Lsz - 1 - t) : t;
    v8f ai = {}, af = {}, ag = {}, ao = {};
    for (int k0 = 0; k0 < Hsz; k0 += 4) {
      v2f a;
      a.x = hLDS[lanelo * Hsz + k0 + khalf * 2 + 0];
      a.y = hLDS[lanelo * Hsz + k0 + khalf * 2 + 1];
      const float* w0 = Whh + (size_t)(j +    0) * Hsz + k0 + khalf * 2;
      const float* w1 = Whh + (size_t)(j +  400) * Hsz + k0 + khalf * 2;
      const float* w2 = Whh + (size_t)(j +  800) * Hsz + k0 + khalf * 2;
      const float* w3 = Whh + (size_t)(j + 1200) * Hsz + k0 + khalf * 2;
      v2f bfrag;
      bfrag.x = w0[0]; bfrag.y = w0[1];
      ai = __builtin_amdgcn_wmma_f32_16x16x4_f32(false, a, false, bfrag, (short)0, ai, false, false);
      bfrag.x = w1[0]; bfrag.y = w1[1];
      af = __builtin_amdgcn_wmma_f32_16x16x4_f32(false, a, false, bfrag, (short)0, af, false, false);
      bfrag.x = w2[0]; bfrag.y = w2[1];
      ag = __builtin_amdgcn_wmma_f32_16x16x4_f32(false, a, false, bfrag, (short)0, ag, false, false);
      bfrag.x = w3[0]; bfrag.y = w3[1];
      ao = __builtin_amdgcn_wmma_f32_16x16x4_f32(false, a, false, bfrag, (short)0, ao, false, false);
    }
    __syncthreads();   // everyone finished reading old h
#pragma unroll
    for (int r = 0; r < 8; ++r) {
      int bb = khalf * 8 + r;             // batch row of this C-fragment element
      const float* g = Gp + (size_t)(bb * Lsz + tt) * G4H;
      float xi = ai[r] + g[j];
      float xf = af[r] + g[j + 400];
      float xg = ag[r] + g[j + 800];
      float xo = ao[r] + g[j + 1200];
      float cc = sigmoidf_(xf) * creg[r] + sigmoidf_(xi) * tanhf(xg);
      creg[r] = cc;
      float hh = sigmoidf_(xo) * tanhf(cc);
      hLDS[bb * Hsz + j] = hh;
      xout[(size_t)(bb * Lsz + tt) * XSTRIDE + dir * Hsz + j] = hh;
    }
    __syncthreads();   // new h visible before next step
  }
}

// ------------------------------------------------------------------
// Head factorization: A[b,l,k] = P(x[b,l]) @ W1[:, :800]^T,
//                     Bm[b,l,k] = P(x[b,l]) @ W1[:, 800:]^T
// Polynomial fused into the A-fragment load. N padded 100 -> 112.
// ------------------------------------------------------------------
__global__ void feat_gemm(const float* __restrict__ x,   // (B*L, XSTRIDE) final layer output
                          const float* __restrict__ W1,  // (100, 1600)
                          const float* __restrict__ w1, const float* __restrict__ w2,
                          const float* __restrict__ w3, const float* __restrict__ w4,
                          const float* __restrict__ w5,
                          float* __restrict__ Afeat, float* __restrict__ Bfeat) {
  const int lane   = threadIdx.x & 31;
  const int lanelo = lane & 15;
  const int khalf  = lane >> 4;
  const int mt   = blockIdx.x;   // 0..63
  const int half = blockIdx.y;   // 0 -> Afeat (W1 cols 0:800), 1 -> Bfeat (800:1600)
  const float c1 = w1[0], c2 = w2[0], c3 = w3[0], c4 = w4[0], c5 = w5[0];
  const int row = mt * 16 + lanelo;
  const float* xrow = x + (size_t)row * XSTRIDE;

  v8f acc[7] = {};
  for (int k0 = 0; k0 < 800; k0 += 4) {
    float v0 = xrow[k0 + khalf * 2 + 0];
    float v1 = xrow[k0 + khalf * 2 + 1];
    v2f a;
    a.x = v0 * (c1 + v0 * (c2 + v0 * (c3 + v0 * (c4 + v0 * c5))));
    a.y = v1 * (c1 + v1 * (c2 + v1 * (c3 + v1 * (c4 + v1 * c5))));
#pragma unroll
    for (int i = 0; i < 7; ++i) {
      int col = i * 16 + lanelo;
      v2f bfrag = {};
      if (col < 100) {
        const float* wp = W1 + (size_t)col * G4H + half * 800 + k0 + khalf * 2;
        bfrag.x = wp[0];
        bfrag.y = wp[1];
      }
      acc[i] = __builtin_amdgcn_wmma_f32_16x16x4_f32(false, a, false, bfrag,
                                                     (short)0, acc[i], false, false);
    }
  }
  float* out = half ? Bfeat : Afeat;
#pragma unroll
  for (int i = 0; i < 7; ++i) {
#pragma unroll
    for (int r = 0; r < 8; ++r) {
      int m   = khalf * 8 + r;
      int col = i * 16 + lanelo;
      out[(size_t)(mt * 16 + m) * FEATPAD + col] = acc[i][r];
    }
  }
}

// ------------------------------------------------------------------
// out[l,b,l2] = b2 + sum_k W2[k] * relu(A[b,l,k] + Bm[b,l2,k] + b1[k])
// ------------------------------------------------------------------
__global__ void pair_combine(const float* __restrict__ Afeat, const float* __restrict__ Bfeat,
                             const float* __restrict__ b1, const float* __restrict__ W2,
                             const float* __restrict__ b2, float* __restrict__ out) {
  const int l  = blockIdx.x;
  const int b  = blockIdx.y;
  const int l2 = threadIdx.x;
  __shared__ float arow[100];
  __shared__ float w2s[100];
  for (int k = threadIdx.x; k < 100; k += blockDim.x) {
    arow[k] = Afeat[(size_t)(b * Lsz + l) * FEATPAD + k] + b1[k];
    w2s[k]  = W2[k];
  }
  __syncthreads();
  const float* brow = Bfeat + (size_t)(b * Lsz + l2) * FEATPAD;
  float s = b2[0];
  for (int k = 0; k < 100; ++k) {
    float h = arow[k] + brow[k];
    s += w2s[k] * fmaxf(h, 0.0f);
  }
  out[((size_t)l * Bsz + b) * Lsz + l2] = s;
}

// ------------------------------------------------------------------
extern "C" void kernel_launch(void* const* d_in, const int* in_sizes, int n_in,
                              void* d_out, int out_size, void* d_ws, size_t ws_size,
                              hipStream_t stream) {
  const int*   words = (const int*)d_in[0];
  const int*   pos   = (const int*)d_in[1];
  // d_in[2] = max_length (fixed 64)
  const float* wemb  = (const float*)d_in[3];
  const float* temb  = (const float*)d_in[4];
  const float* Wih0  = (const float*)d_in[5];   // (2,1600,400)
  const float* WihR  = (const float*)d_in[6];   // (4,2,1600,800)
  const float* Whh   = (const float*)d_in[7];   // (5,2,1600,400)
  const float* bih   = (const float*)d_in[8];   // (5,2,1600)
  const float* bhh   = (const float*)d_in[9];   // (5,2,1600)
  const float* w1 = (const float*)d_in[10];
  const float* w2 = (const float*)d_in[11];
  const float* w3 = (const float*)d_in[12];
  const float* w4 = (const float*)d_in[13];
  const float* w5 = (const float*)d_in[14];
  const float* mlpW1 = (const float*)d_in[15];  // (100,1600)
  const float* mlpb1 = (const float*)d_in[16];  // (100,)
  const float* mlpW2 = (const float*)d_in[17];  // (1,100)
  const float* mlpb2 = (const float*)d_in[18];  // (1,)
  float* out = (float*)d_out;

  // Workspace layout (floats): ~20.6 MB total
  float* ws    = (float*)d_ws;
  float* xA    = ws;                                          // 16*64*800
  float* xB    = xA    + (size_t)Bsz * Lsz * XSTRIDE;         // 16*64*800
  float* gbuf  = xB    + (size_t)Bsz * Lsz * XSTRIDE;         // 2*16*64*1600
  float* Afeat = gbuf  + (size_t)2 * Bsz * Lsz * G4H;         // 16*64*112
  float* Bfeat = Afeat + (size_t)Bsz * Lsz * FEATPAD;         // 16*64*112

  embed_kernel<<<(Bsz * Lsz * 400 + 255) / 256, 256, 0, stream>>>(words, pos, wemb, temb, xA);

  float* xcur  = xA;
  float* xnext = xB;
  for (int layer = 0; layer < 5; ++layer) {
    const float* Wih;
    int D;
    if (layer == 0) { Wih = Wih0; D = 400; }
    else            { Wih = WihR + (size_t)(layer - 1) * 2 * G4H * 800; D = 800; }
    lstm_input_proj<<<dim3(64, 25, 2), 32, 0, stream>>>(
        xcur, Wih, bih + (size_t)layer * 2 * G4H, bhh + (size_t)layer * 2 * G4H, gbuf, D);
    lstm_recurrence<<<dim3(2), 800, 0, stream>>>(
        Whh + (size_t)layer * 2 * G4H * Hsz, gbuf, xnext);
    float* tmp = xcur; xcur = xnext; xnext = tmp;
  }

  feat_gemm<<<dim3(64, 2), 32, 0, stream>>>(xcur, mlpW1, w1, w2, w3, w4, w5, Afeat, Bfeat);
  pair_combine<<<dim3(64, 16), 64, 0, stream>>>(Afeat, Bfeat, mlpb1, mlpW2, mlpb2, out);
}